// SpecLossLayer_61418032333461
// MI455X (gfx1250) — compile-verified
//
#include <hip/hip_runtime.h>

// ---- CDNA5 WMMA types -------------------------------------------------------
typedef float v2f __attribute__((ext_vector_type(2)));
typedef float v8f __attribute__((ext_vector_type(8)));
typedef __attribute__((address_space(3))) float lds_float;

#define BMI 64         // macro tile rows (i)
#define BMJ 128        // macro tile rows (j)
#define BK 64          // K chunk
#define NCHUNK 4       // MS / BK
#define LSTRIDE 68     // padded LDS row stride (floats): bank = (4*row + k)%64 -> conflict free
#define MS 256         // samples feature dim
#define NROWS 8192     // N

// ---- zero the two scalar accumulators ---------------------------------------
__global__ void zero_acc_kernel(float* acc) {
    if (threadIdx.x < 2) acc[threadIdx.x] = 0.0f;
}

// ---- per-row squared norms: one wave per row --------------------------------
__global__ __launch_bounds__(256)
void row_sq_kernel(const float* __restrict__ samples, float* __restrict__ sq) {
    int gwave = (blockIdx.x * blockDim.x + threadIdx.x) >> 5;   // row id
    int lane  = threadIdx.x & 31;
    if (gwave >= NROWS) return;
    const float4* p = (const float4*)(samples + (size_t)gwave * MS);
    float s = 0.0f;
    #pragma unroll
    for (int c = lane; c < MS / 4; c += 32) {
        float4 v = p[c];
        s += v.x * v.x + v.y * v.y + v.z * v.z + v.w * v.w;
    }
    #pragma unroll
    for (int off = 16; off > 0; off >>= 1) s += __shfl_down(s, off, 32);
    if (lane == 0) sq[gwave] = s;
}

// ---- fused Gram(WMMA f32) + contrastive-loss epilogue -----------------------
// 64x128 macro tile, 2x2 register tile per wave (1.0 LDS b64 load per WMMA),
// double-buffered K chunks filled by async global->LDS DMA (ASYNCcnt).
__global__ __launch_bounds__(256)
void gram_loss_kernel(const float* __restrict__ samples,
                      const int*   __restrict__ labels,
                      const float* __restrict__ sq,
                      float*       __restrict__ acc) {
    __shared__ __attribute__((aligned(16))) float lsi[2][BMI * LSTRIDE];
    __shared__ __attribute__((aligned(16))) float lsj[2][BMJ * LSTRIDE];
    __shared__ float red[16];

    const int I = blockIdx.y;                // 0..127, i rows [I*64, I*64+63]
    const int J = blockIdx.x;                // 0..63,  j rows [J*128, J*128+127]
    const int i0 = I * BMI;
    const int j0 = J * BMJ;
    if (j0 > i0 + BMI - 1) return;           // entirely above diagonal: exit whole block

    const int t    = threadIdx.x;
    const int wave = t >> 5;
    const int lane = t & 31;
    const int ti0  = (wave >> 2) * 2;        // 0 or 2 : first of two i sub-tiles
    const int tj0  = (wave & 3) * 2;         // 0,2,4,6: first of two j sub-tiles
    const int hf   = lane >> 4;              // lane half selects K pair (0,1) vs (2,3)
    const int l16  = lane & 15;

    // issue 12 async global->LDS b128 DMAs for one 64-wide K chunk into buffer b
    auto issue_fill = [&](int kc, int b) {
        #pragma unroll
        for (int r = 0; r < 12; ++r) {
            int idx = t + 256 * r;           // 0..3071
            int row = idx >> 4;              // 0..191
            int c4  = (idx & 15) << 2;       // 0,4,...,60
            if (r < 4) {                     // rows 0..63 -> lsi
                const float* g = samples + (size_t)(i0 + row) * MS + kc + c4;
                lds_float* d = (lds_float*)&lsi[b][row * LSTRIDE + c4];
                asm volatile("global_load_async_to_lds_b128 %0, %1, off"
                             :: "v"(d), "v"(g) : "memory");
            } else {                         // rows 64..191 -> lsj
                const float* g = samples + (size_t)(j0 + row - BMI) * MS + kc + c4;
                lds_float* d = (lds_float*)&lsj[b][(row - BMI) * LSTRIDE + c4];
                asm volatile("global_load_async_to_lds_b128 %0, %1, off"
                             :: "v"(d), "v"(g) : "memory");
            }
        }
    };

    v8f acc00 = {}, acc01 = {}, acc10 = {}, acc11 = {};

    issue_fill(0, 0);                        // prologue: chunk 0 -> buffer 0

    #pragma unroll
    for (int c = 0; c < NCHUNK; ++c) {
        const int p = c & 1;
        if (c + 1 < NCHUNK) {
            // fills for chunk c+1 overlap the wait + compute of chunk c;
            // buffer 1-p was released by the end-of-compute barrier of iter c-1
            issue_fill((c + 1) * BK, 1 - p);
            asm volatile("s_wait_asynccnt 0xC" ::: "memory");  // newest 12 may remain
        } else {
            asm volatile("s_wait_asynccnt 0x0" ::: "memory");
        }
        __syncthreads();                     // all waves' chunk-c DMAs visible

        #pragma unroll
        for (int kk = 0; kk < BK; kk += 4) {
            int k = kk + hf * 2;
            v2f a0 = *(const v2f*)&lsi[p][(ti0 * 16 + l16) * LSTRIDE + k];
            v2f a1 = *(const v2f*)&lsi[p][((ti0 + 1) * 16 + l16) * LSTRIDE + k];
            v2f b0 = *(const v2f*)&lsj[p][(tj0 * 16 + l16) * LSTRIDE + k];
            v2f b1 = *(const v2f*)&lsj[p][((tj0 + 1) * 16 + l16) * LSTRIDE + k];
            acc00 = __builtin_amdgcn_wmma_f32_16x16x4_f32(false, a0, false, b0,
                                                          (short)0, acc00, false, false);
            acc01 = __builtin_amdgcn_wmma_f32_16x16x4_f32(false, a0, false, b1,
                                                          (short)0, acc01, false, false);
            acc10 = __builtin_amdgcn_wmma_f32_16x16x4_f32(false, a1, false, b0,
                                                          (short)0, acc10, false, false);
            acc11 = __builtin_amdgcn_wmma_f32_16x16x4_f32(false, a1, false, b1,
                                                          (short)0, acc11, false, false);
        }
        __syncthreads();                     // release buffer p for refill (chunk c+2) at iter c+1
    }

    // ---- fused epilogue: S_ij, per-element symmetry weight, label mask, hinge
    const float invM = 1.0f / (float)MS;
    const int jA = j0 + tj0 * 16 + l16;
    const int jB = jA + 16;
    const int   ljA = labels[jA];
    const int   ljB = labels[jB];
    const float sjA = sq[jA];
    const float sjB = sq[jB];

    float t1 = 0.0f, t2 = 0.0f;
    auto contrib = [&](float g, float sqi, int li, int i, float sj, int lj, int j) {
        float S  = (sqi + sj - 2.0f * g) * invM;
        float we = (i > j) ? 2.0f : ((i == j) ? 1.0f : 0.0f);
        if (li == lj) t1 += we * S;
        else { float h = 1.0f - S; if (h > 0.0f) t2 += we * h; }
    };
    #pragma unroll
    for (int v = 0; v < 8; ++v) {
        int iA = i0 + ti0 * 16 + hf * 8 + v; // C/D layout: vgpr v -> M = v + 8*laneHalf
        int iB = iA + 16;
        float sqiA = sq[iA]; int liA = labels[iA];
        float sqiB = sq[iB]; int liB = labels[iB];
        contrib(acc00[v], sqiA, liA, iA, sjA, ljA, jA);
        contrib(acc01[v], sqiA, liA, iA, sjB, ljB, jB);
        contrib(acc10[v], sqiB, liB, iB, sjA, ljA, jA);
        contrib(acc11[v], sqiB, liB, iB, sjB, ljB, jB);
    }

    // ---- wave32 shuffle reduce, then block reduce, then global atomic -------
    #pragma unroll
    for (int off = 16; off > 0; off >>= 1) {
        t1 += __shfl_down(t1, off, 32);
        t2 += __shfl_down(t2, off, 32);
    }
    if (lane == 0) { red[wave] = t1; red[8 + wave] = t2; }
    __syncthreads();
    if (t == 0) {
        float a = 0.0f, b = 0.0f;
        #pragma unroll
        for (int wv = 0; wv < 8; ++wv) { a += red[wv]; b += red[8 + wv]; }
        atomicAdd(&acc[0], a);
        atomicAdd(&acc[1], b);
    }
}

// ---- finalize: L = 10*mean(t1) + 10*mean(t2) --------------------------------
__global__ void finalize_kernel(const float* __restrict__ acc, float* __restrict__ out) {
    if (threadIdx.x == 0) {
        const float invNN = 1.0f / ((float)NROWS * (float)NROWS);
        out[0] = 10.0f * (acc[0] + acc[1]) * invNN;
    }
}

extern "C" void kernel_launch(void* const* d_in, const int* in_sizes, int n_in,
                              void* d_out, int out_size, void* d_ws, size_t ws_size,
                              hipStream_t stream) {
    // setup_inputs order: merged[0], input1[1] (both unused), samples[2], labels[3]
    const float* samples = (const float*)d_in[2];
    const int*   labels  = (const int*)d_in[3];
    float*       out     = (float*)d_out;

    // workspace layout: [0..1] = (t1,t2) accumulators, [2..2+N) = row squared norms
    float* acc = (float*)d_ws;
    float* sq  = acc + 2;

    zero_acc_kernel<<<1, 32, 0, stream>>>(acc);

    // one wave per row: 8192 waves * 32 lanes / 256 threads = 1024 blocks
    row_sq_kernel<<<(NROWS * 32) / 256, 256, 0, stream>>>(samples, sq);

    dim3 grid(NROWS / BMJ, NROWS / BMI);  // 64 x 128; blocks above diagonal exit
    gram_loss_kernel<<<grid, 256, 0, stream>>>(samples, labels, sq, acc);

    finalize_kernel<<<1, 32, 0, stream>>>(acc, out);
}